// CapsNet_44813688766552
// MI455X (gfx1250) — compile-verified
//
#include <hip/hip_runtime.h>
#include <math.h>

// CapsNet forward for MI455X (gfx1250, wave32).
// Pipeline: conv1+relu -> conv2 (f32 WMMA implicit GEMM, double-buffered LDS)
//           -> squash/transpose -> fused u_hat + dynamic routing (b_ij = u_hat*S).

typedef float v2f __attribute__((ext_vector_type(2)));
typedef float v8f __attribute__((ext_vector_type(8)));

#define KTOT 20736   // 256*9*9
#define KP   32      // K panel staged in LDS
#define NP   648     // KTOT / KP
#define LDA  34      // padded LDS row stride (conflict-free for 16x4 frags)
#define MT   64      // M tile (rows = (b,s))
#define NT   32      // N tile (oc)

// ---------------- Kernel 1: conv1 9x9 s1 + ReLU ----------------
// x:(B,1,28,28) w:(256,1,9,9) bias:(256) -> h:(B,256,20,20)
__global__ void conv1_relu_kernel(const float* __restrict__ x,
                                  const float* __restrict__ w,
                                  const float* __restrict__ bias,
                                  float* __restrict__ h) {
    __shared__ float img[784];
    __shared__ float wt[81];
    const int bx = blockIdx.x;
    const int b  = bx >> 8;
    const int oc = bx & 255;
    const int t  = threadIdx.x;
    for (int i = t; i < 784; i += 256) img[i] = x[b * 784 + i];
    if (t < 81) wt[t] = w[oc * 81 + t];
    __syncthreads();
    const float bb = bias[oc];
    for (int o = t; o < 400; o += 256) {
        const int oy = o / 20, ox = o - (o / 20) * 20;
        float acc = bb;
        #pragma unroll
        for (int ky = 0; ky < 9; ++ky)
            #pragma unroll
            for (int kx = 0; kx < 9; ++kx)
                acc = fmaf(img[(oy + ky) * 28 + ox + kx], wt[ky * 9 + kx], acc);
        h[(b * 256 + oc) * 400 + o] = fmaxf(acc, 0.0f);
    }
}

// ---------------- Kernel 2: conv2 9x9 s2 as implicit GEMM (f32 WMMA) -------
// GEMM: C[m=(b,s)][n=oc] = sum_k im2col(h)[m][k] * w2[n][k], M=4608 N=256 K=20736
// 64x32 tile / 128 threads (4 waves, each wave: rows wid*16..+16, both N tiles).
// Double-buffered LDS, register-staged global loads, one barrier per K panel.
// Writes p in capsule layout (B,8,1152): p[b][oc>>5][(oc&31)*36+s], + bias.
__global__ void conv2_wmma_kernel(const float* __restrict__ h,
                                  const float* __restrict__ w,
                                  const float* __restrict__ bias,
                                  float* __restrict__ p) {
    __shared__ float lA[2][MT * LDA];  // [buf][row_local][kk]
    __shared__ float lB[2][NT * LDA];  // [buf][oc_local][kk]
    const int t    = threadIdx.x;      // 0..127
    const int lane = t & 31;
    const int wid  = t >> 5;           // 0..3 : wave's M sub-tile
    const int bm   = blockIdx.x;       // 0..71
    const int bn   = blockIdx.y;       // 0..7

    // ---- loader assignment: kk column per thread, 4-row groups ----
    const int kk_l = t & 31;
    const int rg   = t >> 5;           // 0..3
    // A rows r = rg + 4e (e<16): precompute base addr = b*102400 + ry*20 + rx
    int abase[16];
    #pragma unroll
    for (int e = 0; e < 16; ++e) {
        const int m  = bm * MT + rg + 4 * e;
        const int b  = m / 36, s = m - b * 36;
        const int sy = s / 6,  sx = s - sy * 6;
        abase[e] = b * 102400 + (sy * 2) * 20 + sx * 2;
    }
    // B rows oc_l = rg + 4e (e<8)
    const float* wbase = w + (size_t)(bn * NT + rg) * KTOT + kk_l;

    float av[16], bv[8];
    // ---- stage panel 0 ----
    {
        const int kk  = kk_l;
        const int ic  = kk / 81;
        const int rem = kk - ic * 81;
        const int off = ic * 400 + (rem / 9) * 20 + (rem - (rem / 9) * 9);
        #pragma unroll
        for (int e = 0; e < 16; ++e) av[e] = h[abase[e] + off];
        #pragma unroll
        for (int e = 0; e < 8; ++e)  bv[e] = wbase[(size_t)(4 * e) * KTOT];
    }
    #pragma unroll
    for (int e = 0; e < 16; ++e) lA[0][(rg + 4 * e) * LDA + kk_l] = av[e];
    #pragma unroll
    for (int e = 0; e < 8; ++e)  lB[0][(rg + 4 * e) * LDA + kk_l] = bv[e];

    v8f c0 = {}; v8f c1 = {};
    const int khalf = (lane >> 4) << 1;                 // 0 or 2
    const int arow  = (wid * 16 + (lane & 15)) * LDA;
    const int b0row = (lane & 15) * LDA;
    const int b1row = b0row + 16 * LDA;

    int cur = 0;
    for (int pi = 0; pi < NP; ++pi) {
        __syncthreads();   // buf[cur] filled; buf[cur^1] fully consumed
        if (pi + 1 < NP) { // issue next panel's global loads (overlap w/ WMMA)
            const int kk  = (pi + 1) * KP + kk_l;
            const int ic  = kk / 81;
            const int rem = kk - ic * 81;
            const int ky  = rem / 9;
            const int off = ic * 400 + ky * 20 + (rem - ky * 9);
            #pragma unroll
            for (int e = 0; e < 16; ++e) av[e] = h[abase[e] + off];
            #pragma unroll
            for (int e = 0; e < 8; ++e)  bv[e] = wbase[(size_t)kk - kk_l +
                                                       (size_t)(4 * e) * KTOT];
        }
        const float* A = lA[cur];
        const float* Bt = lB[cur];
        #pragma unroll
        for (int ks = 0; ks < KP; ks += 4) {
            v2f a, f0, f1;
            a.x  = A[arow + ks + khalf];   a.y  = A[arow + ks + khalf + 1];
            f0.x = Bt[b0row + ks + khalf]; f0.y = Bt[b0row + ks + khalf + 1];
            f1.x = Bt[b1row + ks + khalf]; f1.y = Bt[b1row + ks + khalf + 1];
            c0 = __builtin_amdgcn_wmma_f32_16x16x4_f32(false, a, false, f0,
                                                       (short)0, c0, false, false);
            c1 = __builtin_amdgcn_wmma_f32_16x16x4_f32(false, a, false, f1,
                                                       (short)0, c1, false, false);
        }
        if (pi + 1 < NP) { // store staged regs into the other buffer
            const int nb = cur ^ 1;
            #pragma unroll
            for (int e = 0; e < 16; ++e) lA[nb][(rg + 4 * e) * LDA + kk_l] = av[e];
            #pragma unroll
            for (int e = 0; e < 8; ++e)  lB[nb][(rg + 4 * e) * LDA + kk_l] = bv[e];
        }
        cur ^= 1;
    }

    // epilogue: C layout -> m = base + r + 8*(lane>=16), n = base + (lane&15)
    const int mrow0 = bm * MT + wid * 16 + ((lane >> 4) << 3);
    const int oc0   = bn * NT + (lane & 15);
    const int oc1   = oc0 + 16;
    const float bi0 = bias[oc0], bi1 = bias[oc1];
    const int d0 = (oc0 >> 5) * 1152 + (oc0 & 31) * 36;
    const int d1 = (oc1 >> 5) * 1152 + (oc1 & 31) * 36;
    #pragma unroll
    for (int r = 0; r < 8; ++r) {
        const int m = mrow0 + r;
        const int b = m / 36, s = m - b * 36;
        p[b * 9216 + d0 + s] = c0[r] + bi0;
        p[b * 9216 + d1 + s] = c1[r] + bi1;
    }
}

// ---------------- Kernel 3: squash over i (1152) per (b,j), transpose ------
// p:(B,8,1152) -> u:(B,1152,8)   u[b,i,j] = p[b,j,i] * sn/((1+sn)*sqrt(sn))
__global__ void squash_u_kernel(const float* __restrict__ p,
                                float* __restrict__ u) {
    __shared__ float red[128];
    const int bx = blockIdx.x;
    const int b  = bx >> 3, j = bx & 7;
    const int t  = threadIdx.x;
    const float* pr = p + b * 9216 + j * 1152;
    float acc = 0.f;
    for (int i = t; i < 1152; i += 128) { const float v = pr[i]; acc += v * v; }
    red[t] = acc; __syncthreads();
    for (int s = 64; s > 0; s >>= 1) {
        if (t < s) red[t] += red[t + s];
        __syncthreads();
    }
    const float sn = red[0];
    const float scale = sn / ((1.0f + sn) * sqrtf(sn));
    for (int i = t; i < 1152; i += 128)
        u[b * 9216 + i * 8 + j] = pr[i] * scale;
}

// ---------------- Kernel 4: fused u_hat + dynamic routing ------------------
// One block per (b,o). u_hat slice (1152x16) built in LDS (stride-17 padded,
// conflict-free column scans). Key identity: b_ij[b,i,o,j] = u_hat * S[b,o].
// 256 threads = 16 j-columns x 16 scanners; reductions via wave32 shfl_xor.
#define UH_STRIDE 17
__global__ void routing_kernel(const float* __restrict__ u,
                               const float* __restrict__ W,
                               float* __restrict__ out) {
    extern __shared__ float smem[];
    float* uh = smem;                       // [1152][17]
    float* sj = smem + 1152 * UH_STRIDE;    // [16]
    const int bx = blockIdx.x;
    const int b  = bx / 10, o = bx - b * 10;
    const int t  = threadIdx.x;

    // build u_hat[i][j] = sum_k W[i,o,j,k] * u[b,i,k]
    for (int i = t; i < 1152; i += 256) {
        const float* up = u + b * 9216 + i * 8;
        float uk[8];
        #pragma unroll
        for (int k = 0; k < 8; ++k) uk[k] = up[k];
        const float* wp = W + (size_t)(i * 10 + o) * 128;  // 16*8 per (i,o)
        #pragma unroll
        for (int j = 0; j < 16; ++j) {
            float acc = 0.f;
            #pragma unroll
            for (int k = 0; k < 8; ++k) acc = fmaf(wp[j * 8 + k], uk[k], acc);
            uh[i * UH_STRIDE + j] = acc;
        }
    }
    __syncthreads();

    const int j   = t >> 4;   // 0..15 (column)
    const int sub = t & 15;   // 0..15 (scanner within column)
    float S  = 0.f;           // cumulative sum(v_j) scalar => b_ij = uh*S
    float vj = 0.f;
    for (int it = 0; it < 3; ++it) {
        // stable softmax over i of (uh[i][j] * S)
        float m = -3.0e38f;
        for (int i = sub; i < 1152; i += 16)
            m = fmaxf(m, uh[i * UH_STRIDE + j] * S);
        #pragma unroll
        for (int off = 8; off; off >>= 1)
            m = fmaxf(m, __shfl_xor(m, off, 16));
        float esum = 0.f, eusum = 0.f;
        for (int i = sub; i < 1152; i += 16) {
            const float uhv = uh[i * UH_STRIDE + j];
            const float e   = __expf(uhv * S - m);
            esum += e; eusum += e * uhv;
        }
        #pragma unroll
        for (int off = 8; off; off >>= 1) {
            esum  += __shfl_xor(esum,  off, 16);
            eusum += __shfl_xor(eusum, off, 16);
        }
        const float s_j = eusum / esum;       // s_j = sum_i softmax_i * u_hat
        if (sub == 0) sj[j] = s_j;
        __syncthreads();
        float sn = 0.f;
        #pragma unroll
        for (int jj = 0; jj < 16; ++jj) sn += sj[jj] * sj[jj];
        const float scale = sn / ((1.0f + sn) * sqrtf(sn));
        vj = s_j * scale;                     // v_j = squash(s)_j
        float sv = 0.f;
        #pragma unroll
        for (int jj = 0; jj < 16; ++jj) sv += sj[jj] * scale;
        S += sv;                              // b_ij += u_hat * sum(v)
        __syncthreads();
    }
    if (sub == 0) out[(b * 10 + o) * 16 + j] = vj;
}

extern "C" void kernel_launch(void* const* d_in, const int* in_sizes, int n_in,
                              void* d_out, int out_size, void* d_ws, size_t ws_size,
                              hipStream_t stream) {
    const float* x  = (const float*)d_in[0];  // (128,1,28,28)
    const float* w1 = (const float*)d_in[1];  // (256,1,9,9)
    const float* b1 = (const float*)d_in[2];  // (256)
    const float* w2 = (const float*)d_in[3];  // (256,256,9,9)
    const float* b2 = (const float*)d_in[4];  // (256)
    const float* W  = (const float*)d_in[5];  // (1152,10,16,8)
    float* out = (float*)d_out;               // (128,10,16)

    float* h = (float*)d_ws;                  // 128*256*400  = 13,107,200 f
    float* p = h + 13107200;                  // 128*8*1152   =  1,179,648 f
    float* u = p + 1179648;                   // 128*1152*8   =  1,179,648 f
    // total workspace ~59 MB

    conv1_relu_kernel<<<128 * 256, 256, 0, stream>>>(x, w1, b1, h);
    conv2_wmma_kernel<<<dim3(72, 8), 128, 0, stream>>>(h, w2, b2, p);
    squash_u_kernel<<<128 * 8, 128, 0, stream>>>(p, u);
    const size_t lds = (size_t)(1152 * UH_STRIDE + 16) * sizeof(float);
    routing_kernel<<<128 * 10, 256, lds, stream>>>(u, W, out);
}